// LinearUpscaler_455266533721
// MI455X (gfx1250) — compile-verified
//
#include <hip/hip_runtime.h>
#include <hip/hip_bf16.h>
#include <stdint.h>

// Problem dims
#define VOCAB 100000
#define EMBED 128
#define BATCH 64
#define SEQ   200
#define KIDS  50
#define NPOS  (BATCH * SEQ)   // 12800 positions

typedef uint32_t uint32x4 __attribute__((ext_vector_type(4)));
typedef int      int32x8  __attribute__((ext_vector_type(8)));
typedef int      int32x4  __attribute__((ext_vector_type(4)));

// ---------------------------------------------------------------------------
// Kernel 1: transpose W[E=128, V=100000] -> WT[V, E] (LDS tiled, coalesced
// on both the read (along v) and write (along e) sides).
// ---------------------------------------------------------------------------
__global__ __launch_bounds__(256) void transpose_w_kernel(
    const float* __restrict__ W, float* __restrict__ WT) {
  __shared__ float tile[32][33];           // +1 pad: no LDS bank conflicts
  const int v0 = blockIdx.x * 32;          // 3125 tiles along V (100000/32)
  const int e0 = blockIdx.y * 32;          // 4 tiles along E (128/32)
  const int tx = threadIdx.x;              // 0..31
  const int ty = threadIdx.y;              // 0..7

#pragma unroll
  for (int j = 0; j < 4; ++j) {
    const int el = ty + 8 * j;
    tile[el][tx] = W[(size_t)(e0 + el) * VOCAB + (v0 + tx)];
  }
  __syncthreads();
#pragma unroll
  for (int j = 0; j < 4; ++j) {
    const int vl = ty + 8 * j;
    WT[(size_t)(v0 + vl) * EMBED + (e0 + tx)] = tile[tx][vl];
  }
}

// ---------------------------------------------------------------------------
// TDM gather-mode descriptor issue: DMA up to 8 rows (32-bit indices) of
// WT[V,128] f32 (512 B/row) from global into LDS at lds_off.
// D# bit layout per CDNA5 ISA.
// This toolchain's builtin takes 6 args:
//   (uint32x4 g0, int32x8 g1, int32x4, int32x4, int32x8, i32 cpol)
// We fill both the int32x4 pair (G2, G3) and the int32x8 (G2:G3 combined)
// with the eight 32-bit row indices.
// ---------------------------------------------------------------------------
__device__ __forceinline__ void tdm_gather_rows(uint32_t lds_off,
                                                uint64_t gbase,
                                                const int* idx8,
                                                int nvalid) {
  // Group 0 (128b): count=1 | gather_index_size(32b)=1<<30 | gather_mode=1<<31,
  //                 lds_addr, global_addr[56:0], type=2 in [127:126]
  uint32x4 g0 = { 1u | (1u << 30) | (1u << 31),
                  lds_off,
                  (uint32_t)gbase,
                  (uint32_t)(gbase >> 32) | (2u << 30) };
  // Group 1 (256b):
  //  w0: data_size=2 (4B) at [17:16]
  //  w1: tensor_dim0[15:0]=128 at [63:48]
  //  w2: tensor_dim1[15:0]=0x86A0 (100000) at [95:80]
  //  w3: tensor_dim1[31:16]=1, tile_dim0=128 at [127:112]
  //  w4: tile_dim1 = #valid indices at [143:128]
  //  w5: tensor_dim0_stride = 128 at [191:160]
  int32x8 g1 = { (int)(2u << 16),
                 (int)(128u << 16),
                 (int)(0x86A0u << 16),
                 (int)(1u | (128u << 16)),
                 nvalid,
                 128,
                 0,
                 0 };
  // Groups 2/3: eight 32-bit row indices
  int32x4 g2 = { idx8[0], idx8[1], idx8[2], idx8[3] };
  int32x4 g3 = { idx8[4], idx8[5], idx8[6], idx8[7] };
  int32x8 g23 = { idx8[0], idx8[1], idx8[2], idx8[3],
                  idx8[4], idx8[5], idx8[6], idx8[7] };
  __builtin_amdgcn_tensor_load_to_lds(g0, g1, g2, g3, g23, 0);
}

// ---------------------------------------------------------------------------
// Kernel 2: one wave32 per (b,s) position. 7 chunks of (8,8,8,8,8,8,2) ids.
// Double-buffered TDM pipeline on TENSORcnt; each lane owns 4 embed floats.
// ---------------------------------------------------------------------------
__global__ __launch_bounds__(256) void pool_gather_kernel(
    const int* __restrict__ ids, const float* __restrict__ WT,
    const float* __restrict__ bias, float* __restrict__ out) {
  // Per-wave staging: 2 buffers x 8 rows x 128 f32 = 8 KB; 8 waves = 64 KB.
  __shared__ __attribute__((aligned(16))) float stg[8][2][8][EMBED];

  const int wave = __builtin_amdgcn_readfirstlane((int)(threadIdx.x >> 5));
  const int lane = (int)(threadIdx.x & 31);
  const int p    = (int)blockIdx.x * 8 + wave;          // grid sized exactly

  const int* idp = ids + (size_t)p * KIDS;
  const uint64_t wt_base = (uint64_t)(uintptr_t)WT;
  const uint32_t lds_b[2] = {
      (uint32_t)(uintptr_t)&stg[wave][0][0][0],
      (uint32_t)(uintptr_t)&stg[wave][1][0][0] };

  float4 acc = *(const float4*)(bias + lane * 4);

  int cur[8], nxt[8];
#pragma unroll
  for (int j = 0; j < 8; ++j) cur[j] = idp[j];
  tdm_gather_rows(lds_b[0], wt_base, cur, 8);

#pragma unroll
  for (int c = 0; c < 7; ++c) {
    const int n = (c == 6) ? 2 : 8;
    if (c < 6) {
      const int base = (c + 1) * 8;
#pragma unroll
      for (int j = 0; j < 8; ++j) {
        const int kk = base + j;
        nxt[j] = (kk < KIDS) ? idp[kk] : 0;
      }
      tdm_gather_rows(lds_b[(c + 1) & 1], wt_base, nxt,
                      (c == 5) ? 2 : 8);
      __builtin_amdgcn_s_wait_tensorcnt(1);   // chunk c has landed in LDS
    } else {
      __builtin_amdgcn_s_wait_tensorcnt(0);   // drain tail
    }

    const float4* buf = (const float4*)&stg[wave][c & 1][0][0];
#pragma unroll
    for (int j = 0; j < 8; ++j) {
      if (j < n && cur[j] != 0) {             // id==0 column is masked out
        float4 r = buf[j * 32 + lane];        // ds_load_b128, conflict-free
        acc.x += r.x; acc.y += r.y; acc.z += r.z; acc.w += r.w;
      }
    }
#pragma unroll
    for (int j = 0; j < 8; ++j) cur[j] = nxt[j];
  }

  *(float4*)(out + (size_t)p * EMBED + lane * 4) = acc;  // coalesced b128
}

// ---------------------------------------------------------------------------
// Launch: inputs = { content_input(int)[64,200,50], W(f32)[128,100000],
//                    b(f32)[128] }; out = f32 [64,200,128]; WT lives in d_ws.
// ---------------------------------------------------------------------------
extern "C" void kernel_launch(void* const* d_in, const int* in_sizes, int n_in,
                              void* d_out, int out_size, void* d_ws, size_t ws_size,
                              hipStream_t stream) {
  const int*   ids  = (const int*)d_in[0];
  const float* W    = (const float*)d_in[1];
  const float* bias = (const float*)d_in[2];
  float*       out  = (float*)d_out;
  float*       WT   = (float*)d_ws;           // needs VOCAB*EMBED*4 = 51.2 MB

  dim3 tgrid(VOCAB / 32, EMBED / 32);         // 3125 x 4
  dim3 tblk(32, 8);
  transpose_w_kernel<<<tgrid, tblk, 0, stream>>>(W, WT);

  dim3 pgrid(NPOS / 8);                       // 1600 blocks x 8 waves
  dim3 pblk(256);
  pool_gather_kernel<<<pgrid, pblk, 0, stream>>>(ids, WT, bias, out);
}